// PlackettLuceLoss_43344809951657
// MI455X (gfx1250) — compile-verified
//
#include <hip/hip_runtime.h>
#include <stdint.h>

#define WAVES_PER_BLOCK 8
#define ROWS_PER_WAVE   8
#define ROWS_PER_BLOCK  (WAVES_PER_BLOCK * ROWS_PER_WAVE)   // 64 rows / block
#define NITEMS          32
#define NEGVAL          (-1.0e4f)

typedef __attribute__((ext_vector_type(2))) float v2f;
typedef __attribute__((ext_vector_type(8))) float v8f;

#define AS1 __attribute__((address_space(1)))
#define AS3 __attribute__((address_space(3)))

// ---- gfx1250 async global->LDS copies (ASYNCcnt-tracked) -------------------
__device__ __forceinline__ void async_g2l_b32(void* lds, const void* gptr) {
#if __has_builtin(__builtin_amdgcn_global_load_async_to_lds_b32)
  __builtin_amdgcn_global_load_async_to_lds_b32(
      (AS1 int*)gptr, (AS3 int*)lds, 0, 0);
#else
  uint32_t loff = (uint32_t)(size_t)(AS3 void*)lds;
  asm volatile("global_load_async_to_lds_b32 %0, %1, off"
               :: "v"(loff), "v"((uint64_t)(size_t)gptr) : "memory");
#endif
}

__device__ __forceinline__ void async_g2l_b8(void* lds, const void* gptr) {
#if __has_builtin(__builtin_amdgcn_global_load_async_to_lds_b8)
  __builtin_amdgcn_global_load_async_to_lds_b8(
      (AS1 char*)gptr, (AS3 char*)lds, 0, 0);
#else
  uint32_t loff = (uint32_t)(size_t)(AS3 void*)lds;
  asm volatile("global_load_async_to_lds_b8 %0, %1, off"
               :: "v"(loff), "v"((uint64_t)(size_t)gptr) : "memory");
#endif
}

#if __has_builtin(__builtin_amdgcn_s_wait_asynccnt)
#define WAIT_ASYNC(n) __builtin_amdgcn_s_wait_asynccnt(n)
#else
#define WAIT_ASYNC(n) asm volatile("s_wait_asynccnt %0" :: "i"(n) : "memory")
#endif

// ---------------------------------------------------------------------------
// Kernel 1: one wave32 per row. Async double-buffered staging of
// (scores, ranks, mask) into LDS, lane-parallel stable sort-by-rank,
// suffix logsumexp scan, per-row loss; per-block partial sum + row count.
// ---------------------------------------------------------------------------
__global__ __launch_bounds__(256) void pl_rows_kernel(
    const float* __restrict__ scores,
    const int* __restrict__ ranks,
    const unsigned char* __restrict__ mask,
    float* __restrict__ part_sum,
    float* __restrict__ part_cnt,
    int nrows)
{
  __shared__ float          lds_s[WAVES_PER_BLOCK][2][NITEMS];
  __shared__ int            lds_r[WAVES_PER_BLOCK][2][NITEMS];
  __shared__ unsigned char  lds_m[WAVES_PER_BLOCK][2][NITEMS];
  __shared__ float          red_s[WAVES_PER_BLOCK];
  __shared__ float          red_c[WAVES_PER_BLOCK];

  const int lane = threadIdx.x & 31;
  const int w    = threadIdx.x >> 5;
  const long long row0 = (long long)blockIdx.x * ROWS_PER_BLOCK
                       + (long long)w * ROWS_PER_WAVE;

  float loss_acc = 0.0f;
  float cnt_acc  = 0.0f;

  auto prefetch = [&](long long row, int buf) {
    const long long base = row * NITEMS + lane;
    async_g2l_b32(&lds_s[w][buf][lane], scores + base);
    async_g2l_b32(&lds_r[w][buf][lane], ranks  + base);
    async_g2l_b8 (&lds_m[w][buf][lane], mask   + base);
  };

  if (row0 < nrows) prefetch(row0, 0);

  int buf = 0;
  for (int i = 0; i < ROWS_PER_WAVE; ++i) {
    const long long row = row0 + i;
    const bool more = (i + 1 < ROWS_PER_WAVE) && (row + 1 < nrows);
    if (more) {
      prefetch(row + 1, buf ^ 1);
      WAIT_ASYNC(3);                 // drain the buffer we are about to read
    } else {
      WAIT_ASYNC(0);
    }

    if (row < nrows) {
      const float s = lds_s[w][buf][lane];
      const int   r = lds_r[w][buf][lane];
      const int   m = (int)lds_m[w][buf][lane];

      // n = number of valid (unmasked) items in this row
      const int n = __popcll(__ballot(m == 0));

      // stable sort position by key (masked -> pushed to end)
      const int key = m ? (NITEMS + 1) : r;
      int pos = 0;
      #pragma unroll
      for (int j = 0; j < 32; ++j) {
        const int kj = __shfl(key, j, 32);
        pos += (int)((kj < key) | ((kj == key) & (j < lane)));
      }

      // scatter: lane `pos` receives this lane's score (wave permutation)
      const float ssort = __int_as_float(
          __builtin_amdgcn_ds_permute(pos << 2, __float_as_int(s)));

      const float sm = (lane < n) ? ssort : NEGVAL;

      // row max (stabilizer for logsumexp)
      float M = sm;
      #pragma unroll
      for (int off = 16; off; off >>= 1) M = fmaxf(M, __shfl_xor(M, off, 32));

      // suffix inclusive sum of exp(sm - M)  (exp(NEGVAL - M) underflows to 0)
      const float e = __expf(sm - M);
      float suf = e;
      #pragma unroll
      for (int off = 1; off < 32; off <<= 1) {
        const float t = __shfl_down(suf, off, 32);
        if (lane + off < 32) suf += t;
      }

      // term_k = s_k - logsumexp(s_{k:}) for k < n-1
      const float term = (lane < n - 1) ? (sm - (M + __logf(suf))) : 0.0f;
      float tsum = term;
      #pragma unroll
      for (int off = 16; off; off >>= 1) tsum += __shfl_xor(tsum, off, 32);

      if (n >= 2) {
        loss_acc += -tsum / (float)n;
        cnt_acc  += 1.0f;
      }
    }
    buf ^= 1;
  }

  if (lane == 0) { red_s[w] = loss_acc; red_c[w] = cnt_acc; }
  __syncthreads();
  if (threadIdx.x == 0) {
    float bs = 0.0f, bc = 0.0f;
    #pragma unroll
    for (int k = 0; k < WAVES_PER_BLOCK; ++k) { bs += red_s[k]; bc += red_c[k]; }
    part_sum[blockIdx.x] = bs;
    part_cnt[blockIdx.x] = bc;
  }
}

// ---------------------------------------------------------------------------
// Kernel 2: single wave32 folds the per-block partials with chained
// V_WMMA_F32_16X16X4_F32 (A = all-ones 16x4 => each WMMA reduces 64 floats;
// 4 independent accumulators hide the accumulate-chain latency).
// nb must be a multiple of 128 (padded region zeroed by host-side memset).
// ---------------------------------------------------------------------------
__global__ __launch_bounds__(32) void pl_reduce_kernel(
    const float* __restrict__ part_sum,
    const float* __restrict__ part_cnt,
    float* __restrict__ out,
    int nb)
{
  const int lane = threadIdx.x;
  float total_sum, total_cnt;

#if __has_builtin(__builtin_amdgcn_wmma_f32_16x16x4_f32)
  v2f a; a.x = 1.0f; a.y = 1.0f;       // all-ones A matrix (16x4)
  v8f cs0 = {}; v8f cs1 = {};
  v8f cc0 = {}; v8f cc1 = {};
  for (int i = 0; i < nb; i += 128) {
    v2f bs0; bs0.x = part_sum[i +   0 + lane]; bs0.y = part_sum[i +  32 + lane];
    v2f bs1; bs1.x = part_sum[i +  64 + lane]; bs1.y = part_sum[i +  96 + lane];
    v2f bc0; bc0.x = part_cnt[i +   0 + lane]; bc0.y = part_cnt[i +  32 + lane];
    v2f bc1; bc1.x = part_cnt[i +  64 + lane]; bc1.y = part_cnt[i +  96 + lane];
    cs0 = __builtin_amdgcn_wmma_f32_16x16x4_f32(false, a, false, bs0, (short)0, cs0, false, false);
    cs1 = __builtin_amdgcn_wmma_f32_16x16x4_f32(false, a, false, bs1, (short)0, cs1, false, false);
    cc0 = __builtin_amdgcn_wmma_f32_16x16x4_f32(false, a, false, bc0, (short)0, cc0, false, false);
    cc1 = __builtin_amdgcn_wmma_f32_16x16x4_f32(false, a, false, bc1, (short)0, cc1, false, false);
  }
  // Row M=0 of D lives in c[0] on lanes 0..15 (col sums); reduce across cols.
  float vs = cs0[0] + cs1[0];
  float vc = cc0[0] + cc1[0];
  #pragma unroll
  for (int off = 8; off; off >>= 1) {
    vs += __shfl_xor(vs, off, 16);
    vc += __shfl_xor(vc, off, 16);
  }
  total_sum = vs;
  total_cnt = vc;
#else
  float vs = 0.0f, vc = 0.0f;
  for (int i = lane; i < nb; i += 32) { vs += part_sum[i]; vc += part_cnt[i]; }
  #pragma unroll
  for (int off = 16; off; off >>= 1) {
    vs += __shfl_xor(vs, off, 32);
    vc += __shfl_xor(vc, off, 32);
  }
  total_sum = vs;
  total_cnt = vc;
#endif

  if (lane == 0) out[0] = total_sum / fmaxf(total_cnt, 1.0f);
}

// ---------------------------------------------------------------------------
extern "C" void kernel_launch(void* const* d_in, const int* in_sizes, int n_in,
                              void* d_out, int out_size, void* d_ws, size_t ws_size,
                              hipStream_t stream)
{
  (void)n_in; (void)out_size; (void)ws_size;
  const float*         scores = (const float*)d_in[0];
  const int*           ranks  = (const int*)d_in[1];   // jnp int (x64 off -> i32)
  const unsigned char* mask   = (const unsigned char*)d_in[2];

  const int total   = in_sizes[0];
  const int nrows   = total / NITEMS;                      // 262144
  const int nblocks = (nrows + ROWS_PER_BLOCK - 1) / ROWS_PER_BLOCK;  // 4096
  const int nb      = ((nblocks + 127) / 128) * 128;       // WMMA tile multiple

  float* part_sum = (float*)d_ws;
  float* part_cnt = part_sum + nb;

  // zero the (padded) partial arrays; graph-capture safe
  (void)hipMemsetAsync(d_ws, 0, (size_t)(2 * nb) * sizeof(float), stream);

  pl_rows_kernel<<<dim3(nblocks), dim3(256), 0, stream>>>(
      scores, ranks, mask, part_sum, part_cnt, nrows);

  pl_reduce_kernel<<<dim3(1), dim3(32), 0, stream>>>(
      part_sum, part_cnt, (float*)d_out, nb);
}